// GNN_24326694764769
// MI455X (gfx1250) — compile-verified
//
#include <hip/hip_runtime.h>

typedef __attribute__((ext_vector_type(2))) float v2f;
typedef __attribute__((ext_vector_type(8))) float v8f;

#define DD 128
#define HH 256
#define LDA (DD + 4)    // padded k-stride for gemm1 (4-bank skew -> conflict free)
#define LDA2 (HH + 4)   // padded k-stride for gemm2
#define RT 8            // row tiles (of 16 rows) per GEMM block

// ---------------- atom encoder: h[n] = sum_f atom_emb[f][x[n,f]] ----------------
__global__ void atom_encode_kernel(const int* __restrict__ x,
                                   const float* __restrict__ atom_emb,
                                   float* __restrict__ h, int Nn) {
  int node = blockIdx.x * 8 + (threadIdx.x >> 5);
  if (node >= Nn) return;
  int c = (threadIdx.x & 31) * 4;
  float4 acc = make_float4(0.f, 0.f, 0.f, 0.f);
#pragma unroll
  for (int f = 0; f < 9; ++f) {
    int v = x[node * 9 + f];
    const float4 e = *(const float4*)(atom_emb + (size_t)(f * 120 + v) * DD + c);
    acc.x += e.x; acc.y += e.y; acc.z += e.z; acc.w += e.w;
  }
  *(float4*)(h + (size_t)node * DD + c) = acc;
}

// ---------------- zero scratch ----------------
__global__ void zero_kernel(float* __restrict__ p, size_t n) {
  size_t i = (size_t)blockIdx.x * blockDim.x + threadIdx.x;
  size_t stride = (size_t)gridDim.x * blockDim.x;
  for (; i < n; i += stride) p[i] = 0.f;
}

// ---------------- edge message + scatter-add: aggr[dst] += relu(h[src] + bond_emb) ----------------
__global__ void msg_scatter_kernel(const int* __restrict__ edge_index,
                                   const int* __restrict__ edge_attr,
                                   const float* __restrict__ bond_l,  // [3][6][128]
                                   const float* __restrict__ h,
                                   float* __restrict__ aggr, int E) {
  int e = blockIdx.x * 8 + (threadIdx.x >> 5);
  if (e >= E) return;
  int c = (threadIdx.x & 31) * 4;
  int s = edge_index[e];
  int d = edge_index[E + e];
  float4 acc = *(const float4*)(h + (size_t)s * DD + c);
#pragma unroll
  for (int f = 0; f < 3; ++f) {
    int v = edge_attr[e * 3 + f];
    const float4 b = *(const float4*)(bond_l + (size_t)(f * 6 + v) * DD + c);
    acc.x += b.x; acc.y += b.y; acc.z += b.z; acc.w += b.w;
  }
  float* out = aggr + (size_t)d * DD + c;
  atomicAdd(out + 0, fmaxf(acc.x, 0.f));
  atomicAdd(out + 1, fmaxf(acc.y, 0.f));
  atomicAdd(out + 2, fmaxf(acc.z, 0.f));
  atomicAdd(out + 3, fmaxf(acc.w, 0.f));
}

// ---------------- GEMM1: z1 = ((1+eps)*h + aggr) @ W1 + b1 ; accumulate column stats ----------------
// Dynamic LDS layout: Bs[HH][LDA] (W1 transposed, k-contiguous) then As[16][LDA].
__global__ void __launch_bounds__(256) gin_gemm1_kernel(
    const float* __restrict__ h, const float* __restrict__ aggr,
    const float* __restrict__ W1, const float* __restrict__ b1,
    const float* __restrict__ eps, int layer,
    float* __restrict__ z1, float* __restrict__ colsum, float* __restrict__ colsq,
    int Nn, int rowTiles) {
  extern __shared__ float smem[];
  float* Bs = smem;             // HH * LDA
  float* As = smem + HH * LDA;  // 16 * LDA
  const int t = threadIdx.x;

  // stage whole W1 into LDS, transposed: Bs[col*LDA + k]
  for (int i = t; i < (DD * HH / 4); i += 256) {
    int idx = i * 4;               // W1 row-major: idx = k*HH + col
    int k = idx >> 8;              // /HH
    int col = idx & (HH - 1);
    const float4 w = *(const float4*)(W1 + idx);
    Bs[(col + 0) * LDA + k] = w.x;
    Bs[(col + 1) * LDA + k] = w.y;
    Bs[(col + 2) * LDA + k] = w.z;
    Bs[(col + 3) * LDA + k] = w.w;
  }
  const float ope = 1.0f + eps[layer];
  const int lane = t & 31;
  const int wave = t >> 5;
  const int laneM = lane & 15;
  const int kOff  = (lane >> 4) << 1;   // 0 or 2
  const int rbase = (lane >> 4) << 3;   // 0 or 8
  __syncthreads();

  for (int rt = 0; rt < RT; ++rt) {
    const int tile = blockIdx.x * RT + rt;
    if (tile >= rowTiles) break;        // uniform across block
    const int row0 = tile * 16;
    // cooperative A-tile build: 16 rows x 128 K, GIN combine fused
    for (int i = t; i < (16 * DD / 4); i += 256) {
      int idx = i * 4;
      int r = idx >> 7;
      int k = idx & (DD - 1);
      int row = row0 + r; if (row >= Nn) row = Nn - 1;
      const float4 hv = *(const float4*)(h + (size_t)row * DD + k);
      const float4 av = *(const float4*)(aggr + (size_t)row * DD + k);
      float4 o;
      o.x = ope * hv.x + av.x;
      o.y = ope * hv.y + av.y;
      o.z = ope * hv.z + av.z;
      o.w = ope * hv.w + av.w;
      *(float4*)(As + r * LDA + k) = o;
    }
    __syncthreads();
#pragma unroll
    for (int sub = 0; sub < 2; ++sub) {
      const int col0 = (wave * 2 + sub) * 16;  // 16 col-tiles over H=256
      const int bbase = (col0 + laneM) * LDA;
      v8f acc = {};
      for (int k = 0; k < DD; k += 4) {
        v2f a, b;
        a.x = As[laneM * LDA + k + kOff];
        a.y = As[laneM * LDA + k + kOff + 1];
        b.x = Bs[bbase + k + kOff];
        b.y = Bs[bbase + k + kOff + 1];
        acc = __builtin_amdgcn_wmma_f32_16x16x4_f32(false, a, false, b, (short)0,
                                                    acc, false, false);
      }
      const int col = col0 + laneM;
      const float bias = b1[col];
      float s = 0.f, sq = 0.f;
#pragma unroll
      for (int i = 0; i < 8; ++i) {
        float v = acc[i] + bias;
        int row = row0 + rbase + i;
        if (row < Nn) {
          z1[(size_t)row * HH + col] = v;
          s += v; sq += v * v;
        }
      }
      // lanes L and L+16 own the same column: pair-reduce, halve the atomics
      s  += __shfl_xor(s, 16, 32);
      sq += __shfl_xor(sq, 16, 32);
      if (lane < 16) {
        atomicAdd(&colsum[col], s);
        atomicAdd(&colsq[col], sq);
      }
    }
    __syncthreads();
  }
}

// ---------------- fold BN stats into per-column affine ----------------
__global__ void stats_kernel(const float* __restrict__ sum, const float* __restrict__ sq,
                             const float* __restrict__ g, const float* __restrict__ b,
                             float* __restrict__ scale, float* __restrict__ shift,
                             float invN, int C) {
  int c = blockIdx.x * blockDim.x + threadIdx.x;
  if (c >= C) return;
  float mu = sum[c] * invN;
  float var = sq[c] * invN - mu * mu;
  float s = g[c] * rsqrtf(var + 1e-5f);
  scale[c] = s;
  shift[c] = b[c] - mu * s;
}

// ---------------- GEMM2: z2 = relu(bn1(z1)) @ W2 + b2 ; accumulate column stats ----------------
// Dynamic LDS layout: Bs[DD][LDA2] (W2 transposed) then As[16][LDA2].
__global__ void __launch_bounds__(256) gin_gemm2_kernel(
    const float* __restrict__ z1, const float* __restrict__ scale1,
    const float* __restrict__ shift1,
    const float* __restrict__ W2, const float* __restrict__ b2,
    float* __restrict__ z2, float* __restrict__ colsum, float* __restrict__ colsq,
    int Nn, int rowTiles) {
  extern __shared__ float smem[];
  float* Bs = smem;              // DD * LDA2
  float* As = smem + DD * LDA2;  // 16 * LDA2
  const int t = threadIdx.x;

  // stage whole W2 into LDS, transposed: Bs[col*LDA2 + k]
  for (int i = t; i < (HH * DD / 4); i += 256) {
    int idx = i * 4;               // W2 row-major: idx = k*DD + col
    int k = idx >> 7;              // /DD
    int col = idx & (DD - 1);
    const float4 w = *(const float4*)(W2 + idx);
    Bs[(col + 0) * LDA2 + k] = w.x;
    Bs[(col + 1) * LDA2 + k] = w.y;
    Bs[(col + 2) * LDA2 + k] = w.z;
    Bs[(col + 3) * LDA2 + k] = w.w;
  }
  const int lane = t & 31;
  const int wave = t >> 5;
  const int laneM = lane & 15;
  const int kOff  = (lane >> 4) << 1;
  const int rbase = (lane >> 4) << 3;
  const int col0 = wave * 16;      // 8 col-tiles over D=128, one per wave
  const int bbase = (col0 + laneM) * LDA2;
  __syncthreads();

  for (int rt = 0; rt < RT; ++rt) {
    const int tile = blockIdx.x * RT + rt;
    if (tile >= rowTiles) break;
    const int row0 = tile * 16;
    // cooperative A-tile build: 16 rows x 256 K, BN1+relu fused
    for (int i = t; i < (16 * HH / 4); i += 256) {
      int idx = i * 4;
      int r = idx >> 8;
      int k = idx & (HH - 1);
      int row = row0 + r; if (row >= Nn) row = Nn - 1;
      const float4 zv = *(const float4*)(z1 + (size_t)row * HH + k);
      const float4 sc = *(const float4*)(scale1 + k);
      const float4 sh = *(const float4*)(shift1 + k);
      float4 o;
      o.x = fmaxf(zv.x * sc.x + sh.x, 0.f);
      o.y = fmaxf(zv.y * sc.y + sh.y, 0.f);
      o.z = fmaxf(zv.z * sc.z + sh.z, 0.f);
      o.w = fmaxf(zv.w * sc.w + sh.w, 0.f);
      *(float4*)(As + r * LDA2 + k) = o;
    }
    __syncthreads();
    v8f acc = {};
    for (int k = 0; k < HH; k += 4) {
      v2f a, b;
      a.x = As[laneM * LDA2 + k + kOff];
      a.y = As[laneM * LDA2 + k + kOff + 1];
      b.x = Bs[bbase + k + kOff];
      b.y = Bs[bbase + k + kOff + 1];
      acc = __builtin_amdgcn_wmma_f32_16x16x4_f32(false, a, false, b, (short)0,
                                                  acc, false, false);
    }
    const int col = col0 + laneM;
    const float bias = b2[col];
    float s = 0.f, sq = 0.f;
#pragma unroll
    for (int i = 0; i < 8; ++i) {
      float v = acc[i] + bias;
      int row = row0 + rbase + i;
      if (row < Nn) {
        z2[(size_t)row * DD + col] = v;
        s += v; sq += v * v;
      }
    }
    s  += __shfl_xor(s, 16, 32);
    sq += __shfl_xor(sq, 16, 32);
    if (lane < 16) {
      atomicAdd(&colsum[col], s);
      atomicAdd(&colsq[col], sq);
    }
    __syncthreads();
  }
}

// ---------------- h = relu(bn2(z2)) ----------------
__global__ void bn_relu_kernel(const float* __restrict__ z, const float* __restrict__ scale,
                               const float* __restrict__ shift, float* __restrict__ out,
                               int n4) {
  int i = blockIdx.x * blockDim.x + threadIdx.x;
  if (i >= n4) return;
  int c = (i * 4) & (DD - 1);
  const float4 zv = *(const float4*)(z + (size_t)i * 4);
  const float4 sc = *(const float4*)(scale + c);
  const float4 sh = *(const float4*)(shift + c);
  float4 o;
  o.x = fmaxf(zv.x * sc.x + sh.x, 0.f);
  o.y = fmaxf(zv.y * sc.y + sh.y, 0.f);
  o.z = fmaxf(zv.z * sc.z + sh.z, 0.f);
  o.w = fmaxf(zv.w * sc.w + sh.w, 0.f);
  *(float4*)(out + (size_t)i * 4) = o;
}

extern "C" void kernel_launch(void* const* d_in, const int* in_sizes, int n_in,
                              void* d_out, int out_size, void* d_ws, size_t ws_size,
                              hipStream_t stream) {
  const int*   x          = (const int*)d_in[0];
  const int*   edge_index = (const int*)d_in[1];
  const int*   edge_attr  = (const int*)d_in[2];
  const float* atom_emb   = (const float*)d_in[3];
  const float* bond_emb   = (const float*)d_in[4];   // [5][3][6][128]
  const float* W1         = (const float*)d_in[5];   // [5][128][256]
  const float* b1         = (const float*)d_in[6];   // [5][256]
  const float* bn1_g      = (const float*)d_in[7];
  const float* bn1_b      = (const float*)d_in[8];
  const float* W2         = (const float*)d_in[9];   // [5][256][128]
  const float* b2         = (const float*)d_in[10];  // [5][128]
  const float* eps        = (const float*)d_in[11];  // [5]
  const float* out_bn_g   = (const float*)d_in[12];
  const float* out_bn_b   = (const float*)d_in[13];

  const int Nn = in_sizes[0] / 9;
  const int E  = in_sizes[2] / 3;
  const float invN = 1.0f / (float)Nn;
  const int rowTiles = (Nn + 15) / 16;
  const int gemmBlocks = (rowTiles + RT - 1) / RT;
  const size_t lds1 = (size_t)(HH * LDA + 16 * LDA) * sizeof(float);    // ~143.6 KB
  const size_t lds2 = (size_t)(DD * LDA2 + 16 * LDA2) * sizeof(float);  // ~149.8 KB

  float* ws    = (float*)d_ws;
  float* hbuf  = ws;                          // N*128
  float* aggr  = hbuf + (size_t)Nn * DD;      // N*128 (reused as z2)
  float* z1    = aggr + (size_t)Nn * DD;      // N*256
  float* stats = z1 + (size_t)Nn * HH;        // 1536 floats
  float* sum1 = stats, *sq1 = stats + 256, *scale1 = stats + 512, *shift1 = stats + 768;
  float* sum2 = stats + 1024, *sq2 = stats + 1152, *scale2 = stats + 1280, *shift2 = stats + 1408;

  atom_encode_kernel<<<(Nn + 7) / 8, 256, 0, stream>>>(x, atom_emb, hbuf, Nn);

  for (int l = 0; l < 5; ++l) {
    zero_kernel<<<2048, 256, 0, stream>>>(aggr, (size_t)Nn * DD);
    zero_kernel<<<8, 256, 0, stream>>>(stats, 1536);

    msg_scatter_kernel<<<(E + 7) / 8, 256, 0, stream>>>(
        edge_index, edge_attr, bond_emb + (size_t)l * 3 * 6 * DD, hbuf, aggr, E);

    gin_gemm1_kernel<<<gemmBlocks, 256, lds1, stream>>>(
        hbuf, aggr, W1 + (size_t)l * DD * HH, b1 + (size_t)l * HH,
        eps, l, z1, sum1, sq1, Nn, rowTiles);

    stats_kernel<<<1, 256, 0, stream>>>(sum1, sq1, bn1_g + (size_t)l * HH,
                                        bn1_b + (size_t)l * HH, scale1, shift1, invN, HH);

    gin_gemm2_kernel<<<gemmBlocks, 256, lds2, stream>>>(
        z1, scale1, shift1, W2 + (size_t)l * HH * DD, b2 + (size_t)l * DD,
        aggr /* z2 reuses aggr */, sum2, sq2, Nn, rowTiles);

    stats_kernel<<<1, 128, 0, stream>>>(sum2, sq2, out_bn_g + (size_t)l * DD,
                                        out_bn_b + (size_t)l * DD, scale2, shift2, invN, DD);

    float* hout = (l == 4) ? (float*)d_out : hbuf;
    int n4 = (Nn * DD) / 4;
    bn_relu_kernel<<<(n4 + 255) / 256, 256, 0, stream>>>(aggr, scale2, shift2, hout, n4);
  }
}